// DPFPCell_68539088109957
// MI455X (gfx1250) — compile-verified
//
#include <hip/hip_runtime.h>
#include <hip/hip_bf16.h>
#include <math.h>

// ---------------------------------------------------------------------------
// DPFP fast-weight cell for MI455X (gfx1250, wave32).
// B=64, DIM=512, NU=1, INNER=1024.
// Memory-bound: W/W_new are 256MB each. Traffic budget:
//   - 1 fused pass over W for rowsum+colsum          (256 MB read)
//   - 1 pass W read (NT/last-use) + W_new write (NT) (512 MB)
//   => ~768 MB @ 23.3 TB/s ~= 33 us. GEMMs (~0.2 GFLOP) use exact-f32 WMMA.
// ---------------------------------------------------------------------------

#define Bsz   64
#define DIM   512
#define INNER 1024

typedef __attribute__((ext_vector_type(2))) float v2f;
typedef __attribute__((ext_vector_type(4))) float v4f;
typedef __attribute__((ext_vector_type(8))) float v8f;

// ---------------------------------------------------------------------------
// One 16x16 output tile of C = A * B^T + bias, fp32, via V_WMMA_F32_16X16X4_F32.
// A: MxK row-major (lda), Bm: NxK row-major (ldb) => both operands fetch
// 2 contiguous K floats per lane (8-byte loads).
// A layout: lane L: M = L&15, K-pair = (L>>4)*2 within each K0 step of 4.
// B (4x16) mirrors with N = L&15. C/D: vgpr r -> M = r + 8*(L>>4), N = L&15.
// ---------------------------------------------------------------------------
__device__ __forceinline__ void gemm_nt_tile_f32(
    const float* __restrict__ A, int lda,
    const float* __restrict__ Bm, int ldb,
    const float* __restrict__ bias,
    float* __restrict__ C, int ldc,
    int M0, int N0, int K, int lane)
{
    const int mrow = lane & 15;
    const int kgrp = (lane >> 4) << 1;   // 0 or 2
    const float* ap = A  + (size_t)(M0 + mrow) * lda + kgrp;
    const float* bp = Bm + (size_t)(N0 + mrow) * ldb + kgrp;

    v8f acc = {};
    #pragma unroll 8
    for (int k0 = 0; k0 < K; k0 += 4) {
        v2f a = *(const v2f*)(ap + k0);
        v2f b = *(const v2f*)(bp + k0);
        acc = __builtin_amdgcn_wmma_f32_16x16x4_f32(
            /*neg_a=*/false, a, /*neg_b=*/false, b,
            /*c_mod=*/(short)0, acc, /*reuse_a=*/false, /*reuse_b=*/false);
    }

    const int crow = (lane >> 4) * 8;    // 0 or 8
    const int ccol = lane & 15;
    const float bv = bias[N0 + ccol];
    #pragma unroll
    for (int r = 0; r < 8; ++r)
        C[(size_t)(M0 + crow + r) * ldc + N0 + ccol] = acc[r] + bv;
}

// ---- 1) q,k,v projections: 512 waves, one 16x16 tile each ------------------
__global__ void proj_kernel(const float* __restrict__ x,
                            const float* __restrict__ Wq, const float* __restrict__ bq,
                            const float* __restrict__ Wk, const float* __restrict__ bk,
                            const float* __restrict__ Wv, const float* __restrict__ bv,
                            float* __restrict__ q, float* __restrict__ k,
                            float* __restrict__ v)
{
    const int wave = (blockIdx.x * blockDim.x + threadIdx.x) >> 5;
    const int lane = threadIdx.x & 31;
    if (wave < 128) {                          // q: 4 x 32 tiles
        const int mt = wave >> 5, nt = wave & 31;
        gemm_nt_tile_f32(x, DIM, Wq, DIM, bq, q, DIM, mt * 16, nt * 16, DIM, lane);
    } else if (wave < 256) {                   // k: 4 x 32 tiles
        const int w = wave - 128;
        const int mt = w >> 5, nt = w & 31;
        gemm_nt_tile_f32(x, DIM, Wk, DIM, bk, k, DIM, mt * 16, nt * 16, DIM, lane);
    } else {                                   // v: 4 x 64 tiles
        const int w = wave - 256;
        const int mt = w >> 6, nt = w & 63;
        gemm_nt_tile_f32(x, DIM, Wv, DIM, bv, v, INNER, mt * 16, nt * 16, DIM, lane);
    }
}

// ---- 2) beta = sigmoid(x @ Wbeta^T + bbeta) --------------------------------
__global__ void beta_kernel(const float* __restrict__ x,
                            const float* __restrict__ Wbeta,
                            const float* __restrict__ bbeta,
                            float* __restrict__ beta)
{
    const int b = blockIdx.x * blockDim.x + threadIdx.x;
    if (b < Bsz) {
        float z = bbeta[0];
        for (int c = 0; c < DIM; ++c) z += x[b * DIM + c] * Wbeta[c];
        beta[b] = 1.0f / (1.0f + expf(-z));
    }
}

// ---- 3) dpfp: kp = h(k)*roll(h(k),1), qp likewise --------------------------
__device__ __forceinline__ float hval(const float* __restrict__ base, int b, int j)
{
    float t = (j < DIM) ? base[b * DIM + j] : -base[b * DIM + (j - DIM)];
    return t > 0.0f ? t : 0.0f;
}

__global__ void kpqp_kernel(const float* __restrict__ q, const float* __restrict__ k,
                            float* __restrict__ qp, float* __restrict__ kp)
{
    const int idx = blockIdx.x * blockDim.x + threadIdx.x;   // B*INNER
    const int b = idx >> 10;
    const int j = idx & (INNER - 1);
    const int jm = (j + INNER - 1) & (INNER - 1);
    kp[idx] = hval(k, b, j) * hval(k, b, jm);
    qp[idx] = hval(q, b, j) * hval(q, b, jm);
}

// ---- 4) Skp[b] = sum_j kp[b,j]  (one wave per batch row) -------------------
__global__ void skp_kernel(const float* __restrict__ kp, float* __restrict__ Skp)
{
    const int b = blockIdx.x;
    const int lane = threadIdx.x;
    float s = 0.0f;
    for (int t = lane; t < INNER; t += 32) s += kp[b * INNER + t];
    #pragma unroll
    for (int o = 16; o > 0; o >>= 1) s += __shfl_xor(s, o, 32);
    if (lane == 0) Skp[b] = s;
}

// ---- 5a) zero colsum (atomic accumulation target) --------------------------
__global__ void zero_kernel(float* __restrict__ p)
{
    p[blockIdx.x * blockDim.x + threadIdx.x] = 0.0f;
}

// ---- 5b) fused single pass over W: rowsum (axis=2) + colsum (axis=1) -------
// Block = 256 threads handles batch b, rows [r0, r0+16), all 1024 cols.
// Thread t owns float4 column-group 4t..4t+3: accumulates colsum partials in
// registers across the 16 rows; per-row total via shfl reduce + LDS combine.
__global__ void wsum_kernel(const float* __restrict__ W,
                            float* __restrict__ rowsum,
                            float* __restrict__ colsum)
{
    const int b     = blockIdx.x >> 6;          // /64
    const int chunk = blockIdx.x & 63;
    const int r0    = chunk * 16;
    const int t     = threadIdx.x;              // 0..255 -> float4 col group
    const int lane  = t & 31;
    const int wid   = t >> 5;

    const v4f* base =
        (const v4f*)(W + (size_t)b * INNER * INNER + (size_t)r0 * INNER);

    __shared__ float wred[8][16];               // [wave][row] partials
    v4f cacc = {};

    #pragma unroll
    for (int r = 0; r < 16; ++r) {
        v4f w = base[r * (INNER / 4) + t];
        cacc += w;
        float s = (w.x + w.y) + (w.z + w.w);
        #pragma unroll
        for (int o = 16; o > 0; o >>= 1) s += __shfl_xor(s, o, 32);
        if (lane == 0) wred[wid][r] = s;
    }
    __syncthreads();
    if (t < 16) {
        float rs = 0.f;
        #pragma unroll
        for (int wv = 0; wv < 8; ++wv) rs += wred[wv][t];
        rowsum[(size_t)b * INNER + r0 + t] = rs;
    }
    float* cs = colsum + (size_t)b * INNER + 4 * t;
    atomicAdd(cs + 0, cacc.x);
    atomicAdd(cs + 1, cacc.y);
    atomicAdd(cs + 2, cacc.z);
    atomicAdd(cs + 3, cacc.w);
}

// ---- 6) dv = beta*(v - colsum*kp); readv = (rowsum + dv*Skp)*qp ------------
__global__ void dvread_kernel(const float* __restrict__ v, const float* __restrict__ kp,
                              const float* __restrict__ qp,
                              const float* __restrict__ colsum,
                              const float* __restrict__ rowsum,
                              const float* __restrict__ beta,
                              const float* __restrict__ Skp,
                              float* __restrict__ dv, float* __restrict__ readv)
{
    const int idx = blockIdx.x * blockDim.x + threadIdx.x;   // B*INNER
    const int b = idx >> 10;
    const float d = beta[b] * (v[idx] - colsum[idx] * kp[idx]);
    dv[idx] = d;
    readv[idx] = (rowsum[idx] + d * Skp[b]) * qp[idx];
}

// ---- 7) W_new = W + dv kp^T  (one block per row; 512MB streaming, NT) ------
__global__ void wnew_kernel(const float* __restrict__ W, const float* __restrict__ dv,
                            const float* __restrict__ kp, float* __restrict__ Wnew)
{
    const size_t row = blockIdx.x;                 // b*INNER + i, 65536 rows
    const int    b   = (int)(row >> 10);
    const int    j4  = threadIdx.x;                // 0..255 float4 per row
    const float  d   = dv[row];
    const v4f    kp4 = ((const v4f*)(kp + (size_t)b * INNER))[j4];
    // last-use load of W, non-temporal store of W_new (never re-read)
    const v4f w = __builtin_nontemporal_load((const v4f*)(W + row * INNER) + j4);
    v4f r = w + d * kp4;
    __builtin_nontemporal_store(r, (v4f*)(Wnew + row * INNER) + j4);
}

// ---- 8) out = readv @ Wo^T + bo  (WMMA) ------------------------------------
__global__ void outg_kernel(const float* __restrict__ readv,
                            const float* __restrict__ Wo, const float* __restrict__ bo,
                            float* __restrict__ out)
{
    const int wave = (blockIdx.x * blockDim.x + threadIdx.x) >> 5;  // 128 tiles
    const int lane = threadIdx.x & 31;
    const int mt = wave >> 5, nt = wave & 31;                       // 4 x 32
    gemm_nt_tile_f32(readv, INNER, Wo, INNER, bo, out, DIM, mt * 16, nt * 16, INNER, lane);
}

// ---------------------------------------------------------------------------
extern "C" void kernel_launch(void* const* d_in, const int* in_sizes, int n_in,
                              void* d_out, int out_size, void* d_ws, size_t ws_size,
                              hipStream_t stream)
{
    (void)in_sizes; (void)n_in; (void)out_size; (void)ws_size;

    // setup_inputs order: x, W, Wq, bq, Wk, bk, Wv, bv, Wo, bo, Wbeta, bbeta
    const float* x     = (const float*)d_in[0];
    const float* W     = (const float*)d_in[1];
    const float* Wq    = (const float*)d_in[2];
    const float* bq    = (const float*)d_in[3];
    const float* Wk    = (const float*)d_in[4];
    const float* bk    = (const float*)d_in[5];
    const float* Wv    = (const float*)d_in[6];
    const float* bv    = (const float*)d_in[7];
    const float* Wo    = (const float*)d_in[8];
    const float* bo    = (const float*)d_in[9];
    const float* Wbeta = (const float*)d_in[10];
    const float* bbeta = (const float*)d_in[11];

    // Outputs: out [64,512] then W_new [64,1024,1024], concatenated flat.
    float* out  = (float*)d_out;
    float* Wnew = (float*)d_out + (size_t)Bsz * DIM;

    // Workspace layout (floats)
    float* ws      = (float*)d_ws;
    float* q       = ws;                            // 32768
    float* k       = q      + (size_t)Bsz * DIM;    // 32768
    float* v       = k      + (size_t)Bsz * DIM;    // 65536
    float* kp      = v      + (size_t)Bsz * INNER;  // 65536
    float* qp      = kp     + (size_t)Bsz * INNER;  // 65536
    float* colsum  = qp     + (size_t)Bsz * INNER;  // 65536
    float* rowsum  = colsum + (size_t)Bsz * INNER;  // 65536
    float* dv      = rowsum + (size_t)Bsz * INNER;  // 65536
    float* readv   = dv     + (size_t)Bsz * INNER;  // 65536
    float* beta    = readv  + (size_t)Bsz * INNER;  // 64
    float* Skp     = beta   + Bsz;                  // 64

    // 1) projections (WMMA): 512 waves -> 64 blocks x 256 threads (8 waves)
    proj_kernel<<<64, 256, 0, stream>>>(x, Wq, bq, Wk, bk, Wv, bv, q, k, v);

    // 2) beta
    beta_kernel<<<2, 32, 0, stream>>>(x, Wbeta, bbeta, beta);

    // 3) dpfp features: B*INNER = 65536 threads
    kpqp_kernel<<<256, 256, 0, stream>>>(q, k, qp, kp);

    // 4) Skp: one wave per batch row
    skp_kernel<<<Bsz, 32, 0, stream>>>(kp, Skp);

    // 5) fused W reductions: zero colsum, then single 256MB pass
    zero_kernel<<<256, 256, 0, stream>>>(colsum);
    wsum_kernel<<<Bsz * 64, 256, 0, stream>>>(W, rowsum, colsum);

    // 6) dv + read vector
    dvread_kernel<<<256, 256, 0, stream>>>(v, kp, qp, colsum, rowsum, beta, Skp,
                                           dv, readv);

    // 7) fast-weight update: one block per (b,i) row, float4 NT streaming
    wnew_kernel<<<Bsz * INNER, 256, 0, stream>>>(W, dv, kp, Wnew);

    // 8) output projection (WMMA): 128 waves -> 16 blocks x 256 threads
    outg_kernel<<<16, 256, 0, stream>>>(readv, Wo, bo, out);
}